// TritonGeluD8_45904610460293
// MI455X (gfx1250) — compile-verified
//
#include <hip/hip_runtime.h>

// MI455X / gfx1250, wave32. y = T2 * gelu(T1 * x), 8x8 constant butterflies,
// 12.6M positions. HBM floor ~805MB / 23.3 TB/s ~= 35 us -> must stay BW-bound.
// - Butterflies on the matrix pipe: V_WMMA_F32_16X16X4_F32, two chained K=4
//   WMMAs per transform; row-permuted constant A matrices make stage-1 D regs
//   exactly the stage-2 B layout and stage-2 D regs match output addressing
//   (zero cross-lane moves).
// - GELU via gfx1250 hardware v_tanh_f32 (branchless tanh-GELU, ~6 VALU/val)
//   instead of libm erff (~50 VALU with divergent branches).
// - Scalar (readfirstlane) loop bounds so EXEC stays all-1s for WMMA.
// - Non-temporal loads/stores: 805MB streamed once > 192MB L2.

typedef float v2f __attribute__((ext_vector_type(2)));
typedef float v8f __attribute__((ext_vector_type(8)));

#define C_DIM 384u              // reference C (fixed by harness shapes)

__device__ __forceinline__ float lane_const(unsigned lane, unsigned validMask,
                                            unsigned signMask) {
    unsigned bits = 0x3EB504F3u | (((signMask >> lane) & 1u) << 31); // +/- sqrt(2)/4
    return ((validMask >> lane) & 1u) ? __uint_as_float(bits) : 0.0f;
}

__device__ __forceinline__ float tanh_hw(float z) {
#if __has_builtin(__builtin_amdgcn_tanhf)
    return __builtin_amdgcn_tanhf(z);
#elif __has_builtin(__builtin_amdgcn_tanh_f32)
    return __builtin_amdgcn_tanh_f32(z);
#else
    float r;
    // v_nop after the TRANS op satisfies the 1-op result-use hazard rule
    asm("v_tanh_f32 %0, %1\n\tv_nop" : "=v"(r) : "v"(z));
    return r;
#endif
}

__device__ __forceinline__ float gelu_f(float x) {
    // tanh-form GELU: 0.5*x*(1 + tanh(sqrt(2/pi)*(x + 0.044715*x^3)))
    float u = x * x;
    float z = 0.7978845608028654f * x * __builtin_fmaf(0.044715f, u, 1.0f);
    float t = tanh_hw(z);
    return x * __builtin_fmaf(0.5f, t, 0.5f);
}

__global__ __launch_bounds__(256) void hadamard_gelu_wmma(
    const float* __restrict__ xA1, const float* __restrict__ xA2,
    const float* __restrict__ xB1, const float* __restrict__ xB2,
    const float* __restrict__ x2d, float* __restrict__ out,
    unsigned nA /* = B*N*C */) {

    const unsigned lane = threadIdx.x & 31u;
    const unsigned h    = lane >> 4;     // 0 = lower half-wave, 1 = upper
    const unsigned i    = lane & 15u;    // position within 16-wide tile
    const unsigned nGroups = nA >> 4;    // 16 positions per WMMA tile

    // Force wave id scalar so the loop is a uniform s_cbranch loop
    // (EXEC remains all-1s as WMMA requires).
    const unsigned wid    = (unsigned)__builtin_amdgcn_readfirstlane(threadIdx.x >> 5);
    const unsigned wave   = blockIdx.x * (blockDim.x >> 5) + wid;
    const unsigned nWaves = gridDim.x * (blockDim.x >> 5);

    // Stage-1 A = row-permuted iso_to_reg (m0..m3 = r0,r1,r4,r5; m8..m11 = r2,r3,r6,r7)
    //   -> D vgprs 0..3 = {G0|G2},{G1|G3},{G4|G6},{G5|G7} = stage-2 B layout.
    // Stage-2 A = row-permuted reg_to_iso (m0..m3 = y0,y2,y4,y5; m8..m11 = y1,y3,y6,y7)
    //   -> D vgprs 0..3 = {Y0|Y1},{Y2|Y3},{y4|y6},{y5|y7} = output layout.
    const unsigned VALID = 0x0F0F0F0Fu;
    v2f a1c1 = { lane_const(lane, VALID, 0x0A0A0000u), lane_const(lane, VALID, 0x06060C0Cu) };
    v2f a1c2 = { lane_const(lane, VALID, 0x0906030Cu), lane_const(lane, VALID, 0x000F050Au) };
    v2f a2c1 = { lane_const(lane, VALID, 0x040C0800u), lane_const(lane, VALID, 0x02060E0Au) };
    v2f a2c2 = { lane_const(lane, VALID, 0x03080F04u), lane_const(lane, VALID, 0x0502090Eu) };

    // B-matrix vgpr0 = channels {0|2} = {A1|B1}; vgpr1 = channels {1|3} = {A2|B2}
    const float* __restrict__ pv0 = h ? xB1 : xA1;
    const float* __restrict__ pv1 = h ? xB2 : xA2;

    for (unsigned g = wave; g < nGroups; g += nWaves) {
        const unsigned qbase = g << 4;            // 16-aligned; never straddles bn (16|C)
        const unsigned bn = qbase / C_DIM;        // const divisor -> mul/shift
        const unsigned c  = qbase - bn * C_DIM;
        const unsigned q  = qbase + i;
        // x_2d slot s lives at bn*4C + s*C + c; lower half reads slot s, upper s+1
        const unsigned q2 = bn * (4u * C_DIM) + c + i + h * C_DIM;

        // Streaming loads (non-temporal): two contiguous 64B segments each
        v2f b1 = { __builtin_nontemporal_load(&pv0[q]),
                   __builtin_nontemporal_load(&pv1[q]) };            // {ch0|ch2},{ch1|ch3}
        v2f b2 = { __builtin_nontemporal_load(&x2d[q2]),
                   __builtin_nontemporal_load(&x2d[q2 + 2u*C_DIM]) };// {s0|s1},{s2|s3}

        // Stage 1: G = P1 * x  (K=8 as two chained K=4 WMMAs)
        v8f acc = {};
        acc = __builtin_amdgcn_wmma_f32_16x16x4_f32(false, a1c1, false, b1,
                                                    (short)0, acc, false, false);
        acc = __builtin_amdgcn_wmma_f32_16x16x4_f32(false, a1c2, false, b2,
                                                    (short)0, acc, false, false);

        // GELU on the 4 live D vgprs (all 32 lanes real: 8 ch x 16 pos)
        v2f g01 = { gelu_f(acc[0]), gelu_f(acc[1]) };   // stage-2 B chunk1
        v2f g23 = { gelu_f(acc[2]), gelu_f(acc[3]) };   // stage-2 B chunk2

        // Stage 2: Y = P2 * G  (no shuffles needed)
        v8f y = {};
        y = __builtin_amdgcn_wmma_f32_16x16x4_f32(false, a2c1, false, g01,
                                                  (short)0, y, false, false);
        y = __builtin_amdgcn_wmma_f32_16x16x4_f32(false, a2c2, false, g23,
                                                  (short)0, y, false, false);

        // Streaming stores (outputs concatenated flat: y0,y1,y2,y3 then y_2d)
        float* o01 = out + (h ? nA : 0u);               // {Y0|Y1}
        __builtin_nontemporal_store(y[0], &o01[q]);
        float* o23 = out + (h ? 3u * nA : 2u * nA);     // {Y2|Y3}
        __builtin_nontemporal_store(y[1], &o23[q]);
        const unsigned ob = 4u * nA + q2;               // 2d: slot0<-y4, slot1<-y6
        __builtin_nontemporal_store(y[2], &out[ob]);
        __builtin_nontemporal_store(y[3], &out[ob + 2u * C_DIM]); // slot2<-y5, slot3<-y7
    }
}

extern "C" void kernel_launch(void* const* d_in, const int* in_sizes, int n_in,
                              void* d_out, int out_size, void* d_ws, size_t ws_size,
                              hipStream_t stream) {
    const float* xA1 = (const float*)d_in[0];
    const float* xA2 = (const float*)d_in[1];
    const float* xB1 = (const float*)d_in[2];
    const float* xB2 = (const float*)d_in[3];
    const float* x2d = (const float*)d_in[4];
    float* out = (float*)d_out;

    const unsigned nA = (unsigned)in_sizes[0];     // B*N*C
    const int block = 256;                         // 8 wave32 per block
    const int grid  = 2048;                        // 16384 waves, ~48 tiles each
    hadamard_gelu_wmma<<<grid, block, 0, stream>>>(xA1, xA2, xB1, xB2, x2d, out, nA);
}